// AttentiveFP_12386685682248
// MI455X (gfx1250) — compile-verified
//
#include <hip/hip_runtime.h>
#include <math.h>

typedef __attribute__((ext_vector_type(2))) float v2f;
typedef __attribute__((ext_vector_type(8))) float v8f;

#define NN 300000
#define EE 600000
#define GG 12000
#define NFD 39
#define EFD 11
#define HH 64
#define PP 256
#define CHK 65536

// ---------------------------------------------------------------- helpers
__device__ __forceinline__ float lrelu01(float x) { return x > 0.f ? x : 0.01f * x; }
__device__ __forceinline__ float sigmoidf_(float x) { return 1.f / (1.f + __expf(-x)); }

__device__ __forceinline__ unsigned fkey(float f) {
    unsigned u = __float_as_uint(f);
    return (u & 0x80000000u) ? ~u : (u ^ 0x80000000u);
}
__device__ __forceinline__ float fdec(unsigned k) {
    unsigned u = (k & 0x80000000u) ? (k ^ 0x80000000u) : ~k;
    return __uint_as_float(u);
}

// ---------------------------------------------------------------- fills / packs
__global__ void fill_f32(float* p, float v, long long n) {
    long long t = (long long)blockIdx.x * blockDim.x + threadIdx.x;
    if (t < n) p[t] = v;
}
__global__ void fill_u32(unsigned* p, unsigned v, long long n) {
    long long t = (long long)blockIdx.x * blockDim.x + threadIdx.x;
    if (t < n) p[t] = v;
}
// zero-pad W[K,64] -> out[KP,64]
__global__ void pack_wp(const float* __restrict__ W, float* __restrict__ out, int K,
                        int KP) {
    int t = (int)(blockIdx.x * blockDim.x + threadIdx.x);
    if (t < KP * 64) {
        int k = t >> 6, c = t & 63;
        out[t] = (k < K) ? W[k * 64 + c] : 0.f;
    }
}
// node_feats[N,39] -> xn[N,40] zero-padded
__global__ void pack_xn(const float* __restrict__ nf, float* __restrict__ out) {
    long long t = (long long)blockIdx.x * blockDim.x + threadIdx.x;
    if (t < (long long)NN * 40) {
        int r = (int)(t / 40), c = (int)(t % 40);
        out[t] = (c < NFD) ? nf[(size_t)r * NFD + c] : 0.f;
    }
}
// xe[E,52] = [node_feats[src[e]] (39) | edge_feats[e] (11) | 0 0]
__global__ void pack_xe(const float* __restrict__ nf, const float* __restrict__ ef,
                        const int* __restrict__ src, float* __restrict__ out) {
    long long t = (long long)blockIdx.x * blockDim.x + threadIdx.x;
    if (t < (long long)EE * 52) {
        int r = (int)(t / 52), c = (int)(t % 52);
        float v = 0.f;
        if (c < NFD) v = nf[(size_t)src[r] * NFD + c];
        else if (c < NFD + EFD) v = ef[(size_t)r * EFD + (c - NFD)];
        out[t] = v;
    }
}

// ================================================================ FAST GEMM
// C[M,Nc] = act(A[M,KP] @ W[KP,Nc] + bias),  A contiguous (ld=KP, KP even),
// M % 16 == 0, Nc % 64 == 0.  Wave computes a 16x64 macro-tile; the block's
// KPx64 B-panel is staged in LDS (paired-row layout -> ds_load_b64).
// MODE 0: store.  MODE 1: atomicAdd(Cout[sidx[r]*Nc+c], scale[r]*val)
template <int KP, int ACT, int MODE>
__global__ void gemm_wmma_f(const float* __restrict__ A, const float* __restrict__ W,
                            const float* __restrict__ bias, float* __restrict__ Cout,
                            const float* __restrict__ scale,
                            const int* __restrict__ sidx, int M, int Nc) {
    __shared__ float bsh[(KP / 2) * 128];  // W[k,colbase+c] at (k>>1)*128 + c*2 + (k&1)
    int tilesN4 = Nc >> 6;
    int tn4 = blockIdx.x % tilesN4;
    int bm = blockIdx.x / tilesN4;
    int colbase = tn4 << 6;

#pragma unroll
    for (int i = 0; i < KP / 4; ++i) {  // KP*64 elems / 256 threads
        int e = i * 256 + threadIdx.x;
        int k = e >> 6, cc = e & 63;
        bsh[(k >> 1) * 128 + cc * 2 + (k & 1)] = W[(size_t)k * Nc + colbase + cc];
    }
    __syncthreads();

    int wid = threadIdx.x >> 5;
    int lane = threadIdx.x & 31;
    int hf = lane >> 4, l = lane & 15;
    int tilesM = M >> 4;
    int tm = bm * 8 + wid;
    bool ok = tm < tilesM;
    int tmc = ok ? tm : (tilesM - 1);  // clamp: loads in-bounds, EXEC stays full
    const float* Arow = A + (size_t)(tmc * 16 + l) * KP;

    v8f z = {0.f, 0.f, 0.f, 0.f, 0.f, 0.f, 0.f, 0.f};
    v8f acc0 = z, acc1 = z, acc2 = z, acc3 = z;
#pragma unroll
    for (int k = 0; k < KP; k += 4) {
        int ka = k + 2 * hf;
        v2f a = *(const v2f*)(Arow + ka);
        const float* bb = bsh + (ka >> 1) * 128 + l * 2;
        v2f b0 = *(const v2f*)(bb);
        v2f b1 = *(const v2f*)(bb + 32);
        v2f b2 = *(const v2f*)(bb + 64);
        v2f b3 = *(const v2f*)(bb + 96);
        acc0 = __builtin_amdgcn_wmma_f32_16x16x4_f32(false, a, false, b0, (short)0, acc0, false, false);
        acc1 = __builtin_amdgcn_wmma_f32_16x16x4_f32(false, a, false, b1, (short)0, acc1, false, false);
        acc2 = __builtin_amdgcn_wmma_f32_16x16x4_f32(false, a, false, b2, (short)0, acc2, false, false);
        acc3 = __builtin_amdgcn_wmma_f32_16x16x4_f32(false, a, false, b3, (short)0, acc3, false, false);
    }
    if (!ok) return;  // wave-uniform

    float bc0 = bias[colbase + l];
    float bc1 = bias[colbase + 16 + l];
    float bc2 = bias[colbase + 32 + l];
    float bc3 = bias[colbase + 48 + l];
#pragma unroll
    for (int j = 0; j < 8; ++j) {
        int row = tm * 16 + hf * 8 + j;
        float v0 = acc0[j] + bc0, v1 = acc1[j] + bc1, v2 = acc2[j] + bc2, v3 = acc3[j] + bc3;
        if (ACT == 1) { v0 = lrelu01(v0); v1 = lrelu01(v1); v2 = lrelu01(v2); v3 = lrelu01(v3); }
        if (MODE == 0) {
            float* cp = Cout + (size_t)row * Nc + colbase + l;
            cp[0] = v0; cp[16] = v1; cp[32] = v2; cp[48] = v3;
        } else {
            float sc = scale[row];
            float* cp = Cout + (size_t)sidx[row] * Nc + colbase + l;
            atomicAdd(cp, sc * v0);
            atomicAdd(cp + 16, sc * v1);
            atomicAdd(cp + 32, sc * v2);
            atomicAdd(cp + 48, sc * v3);
        }
    }
}

// ---------------------------------------------------------------- per-row dot (out dim 1)
__global__ void pair_dot(const float* __restrict__ X1, const int* __restrict__ idx1,
                         int relu1, const float* __restrict__ X2,
                         const int* __restrict__ idx2, const float* __restrict__ W,
                         const float* __restrict__ B, float* __restrict__ out, int M) {
    int i = (int)((blockIdx.x * blockDim.x + threadIdx.x) >> 5);
    int lane = threadIdx.x & 31;
    if (i >= M) return;
    int r1 = idx1 ? idx1[i] : i;
    int r2 = idx2 ? idx2[i] : i;
    float2 v1 = ((const float2*)(X1 + (size_t)r1 * HH))[lane];
    float2 v2 = ((const float2*)(X2 + (size_t)r2 * HH))[lane];
    if (relu1) { v1.x = fmaxf(v1.x, 0.f); v1.y = fmaxf(v1.y, 0.f); }
    float2 w1 = ((const float2*)W)[lane];
    float2 w2 = ((const float2*)(W + HH))[lane];
    float p = v1.x * w1.x + v1.y * w1.y + v2.x * w2.x + v2.y * w2.y;
    for (int o = 16; o; o >>= 1) p += __shfl_xor(p, o, 32);
    if (lane == 0) out[i] = lrelu01(p + B[0]);
}

// ---------------------------------------------------------------- segment softmax
__global__ void seg_max_k(const float* __restrict__ x, const int* __restrict__ seg,
                          unsigned* __restrict__ m, int M) {
    int i = (int)(blockIdx.x * blockDim.x + threadIdx.x);
    if (i < M) atomicMax(&m[seg[i]], fkey(x[i]));
}
__global__ void seg_expsum_k(float* __restrict__ x, const int* __restrict__ seg,
                             const unsigned* __restrict__ m, float* __restrict__ s,
                             int M) {
    int i = (int)(blockIdx.x * blockDim.x + threadIdx.x);
    if (i < M) {
        float e = expf(x[i] - fdec(m[seg[i]]));
        x[i] = e;
        atomicAdd(&s[seg[i]], e);
    }
}
__global__ void seg_div_k(float* __restrict__ x, const int* __restrict__ seg,
                          const float* __restrict__ s, int M) {
    int i = (int)(blockIdx.x * blockDim.x + threadIdx.x);
    if (i < M) x[i] = x[i] / s[seg[i]];
}

// ---------------------------------------------------------------- scatter rows (H=64)
__global__ void scatter_rows(const float* __restrict__ X, const int* __restrict__ in_idx,
                             const float* __restrict__ scale,
                             const int* __restrict__ out_idx, float* __restrict__ out,
                             int M) {
    int i = (int)((blockIdx.x * blockDim.x + threadIdx.x) >> 5);
    int lane = threadIdx.x & 31;
    if (i >= M) return;
    int ri = in_idx ? in_idx[i] : i;
    int ro = out_idx[i];
    float s = scale ? scale[i] : 1.f;
    float2 v = ((const float2*)(X + (size_t)ri * HH))[lane];
    atomicAdd(out + (size_t)ro * HH + 2 * lane, s * v.x);
    atomicAdd(out + (size_t)ro * HH + 2 * lane + 1, s * v.y);
}

// ---------------------------------------------------------------- elementwise
__global__ void elu_inplace(float* __restrict__ p, long long n) {
    long long t = (long long)blockIdx.x * blockDim.x + threadIdx.x;
    if (t < n) {
        float x = p[t];
        p[t] = x > 0.f ? x : expm1f(x);
    }
}
__global__ void gru_combine(const float* __restrict__ gi, const float* __restrict__ gh,
                            float* __restrict__ h, int rows) {
    int t = (int)(blockIdx.x * blockDim.x + threadIdx.x);
    if (t >= rows * HH) return;
    int r = t / HH, c = t % HH;
    const float* pi = gi + (size_t)r * 3 * HH;
    const float* ph = gh + (size_t)r * 3 * HH;
    float rr = sigmoidf_(pi[c] + ph[c]);
    float zz = sigmoidf_(pi[HH + c] + ph[HH + c]);
    float nn = tanhf(pi[2 * HH + c] + rr * ph[2 * HH + c]);
    float ho = h[(size_t)r * HH + c];
    float hv = (1.f - zz) * nn + zz * ho;
    h[(size_t)r * HH + c] = fmaxf(hv, 0.f);
}

// ---------------------------------------------------------------- host helpers
static inline int cdiv(long long a, long long b) { return (int)((a + b - 1) / b); }

static void run_gemm64(hipStream_t s, const float* A, const float* W, const float* b,
                       float* C, const float* scale, const int* sidx, int M, int Nc,
                       int mode) {
    int blocks = cdiv(M >> 4, 8) * (Nc >> 6);
    if (mode == 1)
        gemm_wmma_f<64, 0, 1><<<blocks, 256, 0, s>>>(A, W, b, C, scale, sidx, M, Nc);
    else
        gemm_wmma_f<64, 0, 0><<<blocks, 256, 0, s>>>(A, W, b, C, nullptr, nullptr, M, Nc);
}

static void run_softmax(hipStream_t s, float* la, const int* seg, int M, unsigned* segm,
                        float* segs, int nseg) {
    fill_u32<<<cdiv(nseg, 256), 256, 0, s>>>(segm, 0u, nseg);
    fill_f32<<<cdiv(nseg, 256), 256, 0, s>>>(segs, 0.f, nseg);
    seg_max_k<<<cdiv(M, 256), 256, 0, s>>>(la, seg, segm, M);
    seg_expsum_k<<<cdiv(M, 256), 256, 0, s>>>(la, seg, segm, segs, M);
    seg_div_k<<<cdiv(M, 256), 256, 0, s>>>(la, seg, segs, M);
}

static void run_gru(hipStream_t s, const float* x, float* h, int M, const float* Wih,
                    const float* Whh, const float* bih, const float* bhh, float* gi,
                    float* gh) {
    for (int r0 = 0; r0 < M; r0 += CHK) {
        int rows = (M - r0 < CHK) ? (M - r0) : CHK;  // always a multiple of 16 here
        run_gemm64(s, x + (size_t)r0 * HH, Wih, bih, gi, nullptr, nullptr, rows, 3 * HH, 0);
        run_gemm64(s, h + (size_t)r0 * HH, Whh, bhh, gh, nullptr, nullptr, rows, 3 * HH, 0);
        gru_combine<<<cdiv((long long)rows * HH, 256), 256, 0, s>>>(gi, gh,
                                                                    h + (size_t)r0 * HH,
                                                                    rows);
    }
}

// ---------------------------------------------------------------- entry
extern "C" void kernel_launch(void* const* d_in, const int* in_sizes, int n_in,
                              void* d_out, int out_size, void* d_ws, size_t ws_size,
                              hipStream_t stream) {
    (void)in_sizes; (void)n_in; (void)out_size; (void)ws_size;
#define FP(i) ((const float*)d_in[(i)])
    const float* node_feats = FP(0);
    const float* edge_feats = FP(1);
    const int* src = (const int*)d_in[2];
    const int* dst = (const int*)d_in[3];
    const int* node_graph = (const int*)d_in[4];

    char* ws = (char*)d_ws;
    float* h   = (float*)(ws + 0LL);             // [N,64] hv -> h (in place)
    float* he1 = (float*)(ws + 76800000LL);      // [E,64]
    float* pn  = (float*)(ws + 230400000LL);     // [N,64]
    float* ctx = (float*)(ws + 307200000LL);     // [N,64] (also gr [G,64])
    float* av  = (float*)(ws + 384000000LL);     // [E] logits/attn (also [N])
    unsigned* segm = (unsigned*)(ws + 386400000LL);
    float* segs = (float*)(ws + 387600000LL);
    float* gfe = (float*)(ws + 388800000LL);     // [G,64]
    float* gi  = (float*)(ws + 391872000LL);     // [CHK,192]
    float* gh  = (float*)(ws + 442203648LL);     // [CHK,192]
    float* wp1 = (float*)(ws + 492535296LL);     // [40,64] padded project_node W
    float* wp2 = (float*)(ws + 492545536LL);     // [52,64] padded project_edge1 W
    // transient aliases (dead before their hosts are first written):
    float* xn = gi;                              // [N,40]  48MB  (gi region: 50.3MB)
    float* xe = pn;                              // [E,52] 124.8MB (pn+ctx region: 153.6MB)

    // ================= pack inputs =================
    pack_wp<<<cdiv(40 * 64, 256), 256, 0, stream>>>(FP(5), wp1, NFD, 40);
    pack_wp<<<cdiv(52 * 64, 256), 256, 0, stream>>>(FP(7), wp2, NFD + EFD, 52);
    pack_xn<<<cdiv((long long)NN * 40, 256), 256, 0, stream>>>(node_feats, xn);
    pack_xe<<<cdiv((long long)EE * 52, 256), 256, 0, stream>>>(node_feats, edge_feats,
                                                               src, xe);

    // ================= GetContext =================
    gemm_wmma_f<40, 1, 0><<<cdiv(NN >> 4, 8), 256, 0, stream>>>(xn, wp1, FP(6), h,
                                                                nullptr, nullptr, NN, HH);
    gemm_wmma_f<52, 1, 0><<<cdiv(EE >> 4, 8), 256, 0, stream>>>(xe, wp2, FP(8), he1,
                                                                nullptr, nullptr, EE, HH);
    pair_dot<<<cdiv((long long)EE * 32, 256), 256, 0, stream>>>(h, dst, 0, he1, nullptr,
                                                                FP(9), FP(10), av, EE);
    run_softmax(stream, av, dst, EE, segm, segs, NN);
    fill_f32<<<cdiv((long long)NN * HH, 256), 256, 0, stream>>>(ctx, 0.f, (long long)NN * HH);
    run_gemm64(stream, he1, FP(11), FP(12), ctx, av, dst, EE, HH, 1);  // fused scatter
    elu_inplace<<<cdiv((long long)NN * HH, 256), 256, 0, stream>>>(ctx, (long long)NN * HH);
    run_gru(stream, ctx, h, NN, FP(13), FP(14), FP(15), FP(16), gi, gh);

    // ================= GNN layers x2 =================
    for (int l = 0; l < 2; ++l) {
        int b = 17 + 8 * l;
        pair_dot<<<cdiv((long long)EE * 32, 256), 256, 0, stream>>>(h, dst, 0, h, src,
                                                                    FP(b), FP(b + 1), av,
                                                                    EE);
        run_softmax(stream, av, dst, EE, segm, segs, NN);
        run_gemm64(stream, h, FP(b + 2), FP(b + 3), pn, nullptr, nullptr, NN, HH, 0);
        fill_f32<<<cdiv((long long)NN * HH, 256), 256, 0, stream>>>(ctx, 0.f,
                                                                    (long long)NN * HH);
        scatter_rows<<<cdiv((long long)EE * 32, 256), 256, 0, stream>>>(pn, src, av, dst,
                                                                        ctx, EE);
        elu_inplace<<<cdiv((long long)NN * HH, 256), 256, 0, stream>>>(ctx,
                                                                       (long long)NN * HH);
        run_gru(stream, ctx, h, NN, FP(b + 4), FP(b + 5), FP(b + 6), FP(b + 7), gi, gh);
    }

    // ================= Readout =================
    fill_f32<<<cdiv((long long)GG * HH, 256), 256, 0, stream>>>(gfe, 0.f, (long long)GG * HH);
    scatter_rows<<<cdiv((long long)NN * 32, 256), 256, 0, stream>>>(h, nullptr, nullptr,
                                                                    node_graph, gfe, NN);
    for (int t = 0; t < 2; ++t) {
        int b = 33 + 8 * t;
        pair_dot<<<cdiv((long long)NN * 32, 256), 256, 0, stream>>>(gfe, node_graph, 1, h,
                                                                    nullptr, FP(b),
                                                                    FP(b + 1), av, NN);
        run_softmax(stream, av, node_graph, NN, segm, segs, GG);
        run_gemm64(stream, h, FP(b + 2), FP(b + 3), pn, nullptr, nullptr, NN, HH, 0);
        fill_f32<<<cdiv((long long)GG * HH, 256), 256, 0, stream>>>(ctx, 0.f,
                                                                    (long long)GG * HH);
        scatter_rows<<<cdiv((long long)NN * 32, 256), 256, 0, stream>>>(pn, nullptr, av,
                                                                        node_graph, ctx,
                                                                        NN);
        elu_inplace<<<cdiv((long long)GG * HH, 256), 256, 0, stream>>>(ctx,
                                                                       (long long)GG * HH);
        run_gru(stream, ctx, gfe, GG, FP(b + 4), FP(b + 5), FP(b + 6), FP(b + 7), gi, gh);
    }

    run_gemm64(stream, gfe, FP(49), FP(50), (float*)d_out, nullptr, nullptr, GG, PP, 0);
#undef FP
}